// CognitiveModule_76364518523339
// MI455X (gfx1250) — compile-verified
//
#include <hip/hip_runtime.h>

typedef float v2f __attribute__((ext_vector_type(2)));
typedef float v4f __attribute__((ext_vector_type(4)));
typedef float v8f __attribute__((ext_vector_type(8)));
typedef int   v4i __attribute__((ext_vector_type(4)));

#define AS1 __attribute__((address_space(1)))
#define AS3 __attribute__((address_space(3)))

#if defined(__has_builtin)
#if __has_builtin(__builtin_amdgcn_global_load_async_to_lds_b128)
#define USE_ASYNC_LDS 1
#endif
#if __has_builtin(__builtin_amdgcn_s_wait_asynccnt)
#define HAVE_WAIT_ASYNC_BUILTIN 1
#endif
#endif
#ifndef USE_ASYNC_LDS
#define USE_ASYNC_LDS 0
#endif
#ifndef HAVE_WAIT_ASYNC_BUILTIN
#define HAVE_WAIT_ASYNC_BUILTIN 0
#endif

#define DECAY  0.9f
#define THRESH 1.0f

// Layer sizes (from setup_inputs)
constexpr int N1  = 2048;
constexpr int N23 = 8192;
constexpr int N4  = 4096;
constexpr int N56 = 8192;

// Output layout: [s1, s2, s4, s5, v1, v2, v4, v5]
constexpr int OFF_S1 = 0;
constexpr int OFF_S2 = N1;                    // 2048
constexpr int OFF_S4 = N1 + N23;              // 10240
constexpr int OFF_S5 = N1 + N23 + N4;         // 14336
constexpr int OFF_V1 = OFF_S5 + N56;          // 22528
constexpr int OFF_V2 = OFF_V1 + N1;           // 24576
constexpr int OFF_V4 = OFF_V2 + N23;          // 32768
constexpr int OFF_V5 = OFF_V4 + N4;           // 36864

__device__ __forceinline__ void wait_async_then_barrier() {
#if USE_ASYNC_LDS
#if HAVE_WAIT_ASYNC_BUILTIN
  __builtin_amdgcn_s_wait_asynccnt(0);
#else
  asm volatile("s_wait_asynccnt 0" ::: "memory");
#endif
#endif
  __syncthreads();
}

// ---------------------------------------------------------------------------
// Kernel 1: fused elementwise LIF updates for L1, L2_3, L4.
// Writes s2 into d_out so the matvec kernel can read it (stream ordering).
// ---------------------------------------------------------------------------
__global__ __launch_bounds__(256)
void spike_elementwise(const float* __restrict__ ax1,
                       const float* __restrict__ ax2,
                       const float* __restrict__ ext,
                       const float* __restrict__ V1,
                       const float* __restrict__ V2,
                       const float* __restrict__ V4,
                       float* __restrict__ out) {
  int i = blockIdx.x * blockDim.x + threadIdx.x;
  float V, drive;
  int soff, voff, j;
  if (i < N1) {
    j = i;                 V = V1[j]; drive = ax1[j]; soff = OFF_S1; voff = OFF_V1;
  } else if (i < N1 + N23) {
    j = i - N1;            V = V2[j]; drive = ax2[j]; soff = OFF_S2; voff = OFF_V2;
  } else if (i < N1 + N23 + N4) {
    j = i - (N1 + N23);    V = V4[j]; drive = ext[j]; soff = OFF_S4; voff = OFF_V4;
  } else {
    return;
  }
  float Vn = DECAY * V + drive;
  float s  = (Vn >= THRESH) ? 1.0f : 0.0f;
  out[soff + j] = s;
  out[voff + j] = Vn * (1.0f - s);
}

// ---------------------------------------------------------------------------
// Kernel 2: y = W_ff2 @ s2 via V_WMMA_F32_16X16X4_F32, then LIF update for
// L5_6. One block = 16 output rows, 8 waves split the K dimension per chunk.
// W (256 MB, single-use) is streamed HBM -> LDS with async B128 copies
// (ASYNCcnt path, no VGPR landing) when available, NT temporal hint.
// ---------------------------------------------------------------------------
constexpr int KCHUNK = 512;            // K columns staged per iteration
constexpr int LROW   = KCHUNK + 4;     // 516 floats: pad -> conflict-free A reads

__global__ __launch_bounds__(256)
void l56_wmma_matvec(const float* __restrict__ W,      // (N56, N23) row-major
                     const float* __restrict__ s2,     // spikes from L2_3
                     const float* __restrict__ ax56,
                     const float* __restrict__ V56,
                     float* __restrict__ out) {
  __shared__ float tile[16 * LROW];    // 16 rows x 512 K (padded)
  __shared__ float svec[KCHUNK];       // staged spike chunk
  __shared__ float red[8 * 16];        // per-wave partial y

  const int tid  = threadIdx.x;
  const int lane = tid & 31;
  const int wave = tid >> 5;
  const int row0 = blockIdx.x * 16;

  // A-fragment addressing per ISA layout: lanes 0-15 hold K=0,1 for row M=lane;
  // lanes 16-31 hold K=2,3 for row M=lane-16.
  const int r    = lane & 15;
  const int koff = (lane >> 4) << 1;   // 0 or 2

  v8f acc = {0.f, 0.f, 0.f, 0.f, 0.f, 0.f, 0.f, 0.f};

  for (int kc = 0; kc < N23; kc += KCHUNK) {
    // Stage the 512-float spike chunk (threads 0..127, one float4 each).
    if (tid < KCHUNK / 4) {
      v4f sv = *(const v4f*)(s2 + kc + tid * 4);
      *(v4f*)(svec + tid * 4) = sv;
    }
    // Stage the 16x512 W tile: 2048 x 16B, coalesced, non-temporal.
#pragma unroll
    for (int it = 0; it < 8; ++it) {
      int idx = tid + it * 256;        // 0..2047
      int rr  = idx >> 7;              // row within tile (128 float4 per row)
      int c4  = idx & 127;
      const float* gsrc = W + (size_t)(row0 + rr) * N23 + kc + c4 * 4;
      float*       ldst = tile + rr * LROW + c4 * 4;
#if USE_ASYNC_LDS
      // HBM -> LDS direct, no VGPR round-trip; tracked by ASYNCcnt.
      // cpol = 1 -> TH_NT (stream, don't retain in cache).
      __builtin_amdgcn_global_load_async_to_lds_b128(
          (AS1 v4i*)gsrc, (AS3 v4i*)ldst, 0, 1);
#else
      v4f wv = __builtin_nontemporal_load((const v4f*)gsrc);
      *(v4f*)ldst = wv;
#endif
    }
    wait_async_then_barrier();

    // Each wave accumulates its 64-K slice: 16 WMMAs of K=4.
    const int kbase = wave * 64;
#pragma unroll
    for (int kk = 0; kk < 16; ++kk) {
      int k = kbase + kk * 4;
      v2f a = *(const v2f*)(tile + r * LROW + k + koff);   // W[m, k..k+1|k+2..k+3]
      v2f b = *(const v2f*)(svec + k + koff);              // s broadcast over N
      acc = __builtin_amdgcn_wmma_f32_16x16x4_f32(
          /*neg_a=*/false, a, /*neg_b=*/false, b,
          /*c_mod=*/(short)0, acc, /*reuse_a=*/false, /*reuse_b=*/false);
    }
    __syncthreads();
  }

  // D layout: lane 0 holds (M=0..7, N=0); lane 16 holds (M=8..15, N=0).
  // All N columns are identical (B was broadcast), so these two lanes carry y.
  if (lane == 0) {
#pragma unroll
    for (int m = 0; m < 8; ++m) red[wave * 16 + m] = acc[m];
  } else if (lane == 16) {
#pragma unroll
    for (int m = 0; m < 8; ++m) red[wave * 16 + 8 + m] = acc[m];
  }
  __syncthreads();

  if (tid < 16) {
    float sum = 0.f;
#pragma unroll
    for (int w = 0; w < 8; ++w) sum += red[w * 16 + tid];
    int   m  = row0 + tid;
    float Vn = DECAY * V56[m] + (ax56[m] + sum);
    float sp = (Vn >= THRESH) ? 1.0f : 0.0f;
    out[OFF_S5 + m] = sp;
    out[OFF_V5 + m] = Vn * (1.0f - sp);
  }
}

// ---------------------------------------------------------------------------
extern "C" void kernel_launch(void* const* d_in, const int* in_sizes, int n_in,
                              void* d_out, int out_size, void* d_ws, size_t ws_size,
                              hipStream_t stream) {
  (void)in_sizes; (void)n_in; (void)out_size; (void)d_ws; (void)ws_size;

  const float* ext  = (const float*)d_in[0];   // external_input (4096)
  const float* ax1  = (const float*)d_in[1];   // ax_L1   (2048)
  const float* ax2  = (const float*)d_in[2];   // ax_L2_3 (8192)
  const float* ax56 = (const float*)d_in[3];   // ax_L5_6 (8192)
  const float* V1   = (const float*)d_in[4];   // V_L1
  const float* V2   = (const float*)d_in[5];   // V_L2_3
  const float* V4   = (const float*)d_in[6];   // V_L4
  const float* V56  = (const float*)d_in[7];   // V_L5_6
  const float* Wff2 = (const float*)d_in[9];   // W_ff2 (8192 x 8192)
  float* out = (float*)d_out;

  const int nEl = N1 + N23 + N4;               // 14336
  spike_elementwise<<<(nEl + 255) / 256, 256, 0, stream>>>(
      ax1, ax2, ext, V1, V2, V4, out);

  l56_wmma_matvec<<<N56 / 16, 256, 0, stream>>>(
      Wff2, out + OFF_S2, ax56, V56, out);
}